// ScaledDotProductAttention_44976897524605
// MI455X (gfx1250) — compile-verified
//
#include <hip/hip_runtime.h>

// Problem constants (reference: B,H,S,DK = 4,12,1024,64; fp32)
#define BB 4
#define HH 12
#define SS 1024
#define DKK 64
#define QT 16            // query rows per workgroup
#define NWAVES 4         // 4 wave32 per workgroup
#define ROWSTRIDE 1026   // LDS row stride (floats): even (b64-aligned) and 1026%64==2 -> near conflict-free
#define NEG_INF_F (-1.0e9f)
#define SENTINEL_F (-1.0e8f)

typedef float v2f __attribute__((ext_vector_type(2)));
typedef float v8f __attribute__((ext_vector_type(8)));

__global__ __launch_bounds__(128) void
attn_f32_wmma_kernel(const float* __restrict__ Q,
                     const float* __restrict__ K,
                     const float* __restrict__ V,
                     const unsigned char* __restrict__ mask,   // jnp.bool_ -> 1 byte
                     const float* __restrict__ help,
                     float* __restrict__ ctx_out,              // [B,H,S,DK]
                     float* __restrict__ attn_out)             // [B,H,S,S]
{
    extern __shared__ char smem_raw[];
    float* Ssc    = (float*)smem_raw;                 // [QT][ROWSTRIDE] scores -> exp(p)
    float* pred   = Ssc + QT * ROWSTRIDE;             // [16][8] partial max / partial sum
    float* rowMax = pred + 16 * 8;                    // [16]
    float* rowInv = rowMax + 16;                      // [16]

    const int bh    = blockIdx.y;                     // b*H + h
    const int qtile = blockIdx.x;                     // 0..63
    const int qbase = qtile * QT;
    const int tid   = threadIdx.x;
    const int wave  = tid >> 5;
    const int lane  = tid & 31;
    const int hlane = lane & 15;
    const int dsel  = (lane < 16) ? 0 : 2;            // K-offset within a 4-deep WMMA chunk

    const float*         Qp = Q    + (size_t)bh * SS * DKK + (size_t)qbase * DKK;
    const float*         Kp = K    + (size_t)bh * SS * DKK;
    const float*         Vp = V    + (size_t)bh * SS * DKK;
    const unsigned char* Mp = mask + (size_t)bh * SS * SS + (size_t)qbase * SS;
    const float*         Hp = help + (size_t)bh * SS * SS + (size_t)qbase * SS;
    float*               Cp = ctx_out  + (size_t)bh * SS * DKK + (size_t)qbase * DKK;
    float*               Ap = attn_out + (size_t)bh * SS * SS + (size_t)qbase * SS;

    // ---- Preload Q A-fragments (16x4 fp32 A-matrix layout, one float2 per chunk) ----
    // A VGPR0: lanes0-15 K=4c+0, lanes16-31 K=4c+2 ; VGPR1: K=4c+1 / 4c+3 ; M = lane%16
    v2f aq[16];
#pragma unroll
    for (int c = 0; c < 16; ++c) {
        aq[c] = *(const v2f*)(Qp + (size_t)hlane * DKK + 4 * c + dsel);
    }

    // ---- Phase A: scores = (Q K^T)*0.125, masked, blended with help; into LDS ----
    for (int kt = wave; kt < SS / 16; kt += NWAVES) {
        v8f c = {0.f, 0.f, 0.f, 0.f, 0.f, 0.f, 0.f, 0.f};
        // B-matrix 4x16 fp32: VGPR0 lane n -> B[0|2][n] = K[kt*16+n][4cc+dsel]; VGPR1 -> +1
        const float* kb = Kp + (size_t)(kt * 16 + hlane) * DKK + dsel;
#pragma unroll
        for (int cc = 0; cc < 16; ++cc) {
            v2f b = *(const v2f*)(kb + 4 * cc);
            c = __builtin_amdgcn_wmma_f32_16x16x4_f32(
                    false, aq[cc], false, b, (short)0, c, false, false);
        }
        // C layout: VGPR i -> row (lane<16 ? i : 8+i), col = kt*16 + lane%16
        const int col = kt * 16 + hlane;
#pragma unroll
        for (int i = 0; i < 8; ++i) {
            const int row = (lane < 16) ? i : (8 + i);
            float s = c[i] * 0.125f;                        // 1/sqrt(64)
            const size_t g = (size_t)row * SS + col;
            const unsigned char m = __builtin_nontemporal_load(Mp + g);
            if (m) s = NEG_INF_F;
            const float h = __builtin_nontemporal_load(Hp + g);
            if (h > SENTINEL_F) s = 0.5f * h + 0.5f * s;    // valid help: 50/50 blend
            Ssc[row * ROWSTRIDE + col] = s;
        }
    }
    __syncthreads();

    // ---- Softmax over LDS rows (8 threads per row) ----
    {
        const int r = tid >> 3, sl = tid & 7;
        float mx = -INFINITY;
        for (int col = sl; col < SS; col += 8)
            mx = fmaxf(mx, Ssc[r * ROWSTRIDE + col]);
        pred[r * 8 + sl] = mx;
    }
    __syncthreads();
    if (tid < 16) {
        float mx = pred[tid * 8];
#pragma unroll
        for (int j = 1; j < 8; ++j) mx = fmaxf(mx, pred[tid * 8 + j]);
        rowMax[tid] = mx;
    }
    __syncthreads();
    {
        const int r = tid >> 3, sl = tid & 7;
        const float mx = rowMax[r];
        float sum = 0.f;
        for (int col = sl; col < SS; col += 8) {
            const float p = __expf(Ssc[r * ROWSTRIDE + col] - mx);
            Ssc[r * ROWSTRIDE + col] = p;                   // keep unnormalized exp
            sum += p;
        }
        pred[r * 8 + sl] = sum;
    }
    __syncthreads();
    if (tid < 16) {
        float s = 0.f;
#pragma unroll
        for (int j = 0; j < 8; ++j) s += pred[tid * 8 + j];
        rowInv[tid] = 1.0f / s;
    }
    __syncthreads();

    // ---- Stream normalized attention to global (coalesced, non-temporal) ----
    for (int idx = tid; idx < QT * SS; idx += 128) {
        const int r = idx >> 10, col = idx & (SS - 1);
        const float val = Ssc[r * ROWSTRIDE + col] * rowInv[r];
        __builtin_nontemporal_store(val, Ap + (size_t)r * SS + col);
    }

    // ---- Phase B: context = (P @ V) * rowInv ; each wave owns one 16-wide d-tile ----
    const int d0 = wave * 16;
    v8f acc = {0.f, 0.f, 0.f, 0.f, 0.f, 0.f, 0.f, 0.f};
    for (int k0 = 0; k0 < SS; k0 += 4) {
        // A: P[ lane%16 ][ k0+dsel .. k0+dsel+1 ] from LDS (8B-aligned ds_load_b64)
        v2f a = *(const v2f*)&Ssc[hlane * ROWSTRIDE + k0 + dsel];
        // B: V rows k0+dsel, k0+dsel+1 at column d0+lane%16
        const float bx = Vp[(size_t)(k0 + dsel) * DKK + d0 + hlane];
        const float by = Vp[(size_t)(k0 + dsel + 1) * DKK + d0 + hlane];
        v2f b = {bx, by};
        acc = __builtin_amdgcn_wmma_f32_16x16x4_f32(
                  false, a, false, b, (short)0, acc, false, false);
    }
#pragma unroll
    for (int i = 0; i < 8; ++i) {
        const int row = (lane < 16) ? i : (8 + i);
        Cp[(size_t)row * DKK + d0 + hlane] = acc[i] * rowInv[row];
    }
}

extern "C" void kernel_launch(void* const* d_in, const int* in_sizes, int n_in,
                              void* d_out, int out_size, void* d_ws, size_t ws_size,
                              hipStream_t stream) {
    (void)in_sizes; (void)n_in; (void)d_ws; (void)ws_size; (void)out_size;
    const float*         Q    = (const float*)d_in[0];
    const float*         K    = (const float*)d_in[1];
    const float*         V    = (const float*)d_in[2];
    const unsigned char* mask = (const unsigned char*)d_in[3];  // bool mask, 1 byte/elem
    const float*         help = (const float*)d_in[4];

    float* ctx  = (float*)d_out;                                   // [B,H,S,DK]
    float* attn = (float*)d_out + (size_t)BB * HH * SS * DKK;      // [B,H,S,S]

    dim3 grid(SS / QT, BB * HH);   // (64, 48)
    dim3 block(32 * NWAVES);       // 128 threads = 4 wave32
    size_t shmem = (size_t)(QT * ROWSTRIDE + 16 * 8 + 16 + 16) * sizeof(float); // ~66 KB

    attn_f32_wmma_kernel<<<grid, block, shmem, stream>>>(Q, K, V, mask, help, ctx, attn);
}